// Quantize_53017076302344
// MI455X (gfx1250) — compile-verified
//
#include <hip/hip_runtime.h>
#include <hip/hip_bf16.h>

// VQ weight reconstruction: out[i] = centroids[assign[i]] (float4 per entry).
// centroids: 256 x 4 f32 (4 KB), assignments: 8192*2048 ints, out: 8192x8192 f32.
// Pure gather/stream op: ~320 MB of traffic, 0 FLOPs -> HBM-store-bound,
// ~13.8 us floor at 23.3 TB/s. CDNA5 paths: async global->LDS copy (ASYNCcnt),
// ds_load_b128 gathers from LDS, NT-hinted coalesced b128 global stores.
// Guard-free fast path for full blocks keeps 8 loads / 8 stores in flight
// per thread (no s_and_saveexec / per-load s_wait serialization).

typedef float v4f __attribute__((ext_vector_type(4)));  // native vec for NT builtins

#define VQ_K      256   // codebook entries
#define VQ_BLOCK  256   // threads per block (8 wave32)
#define VQ_UNROLL 8     // assignments per thread -> 2048 per block

__global__ __launch_bounds__(VQ_BLOCK) void Quantize_53017076302344_kernel(
    const float* __restrict__ centroids,    // [VQ_K * 4] f32
    const int*   __restrict__ assignments,  // [n] codebook indices
    v4f*         __restrict__ out,          // [n] 16-byte slots
    unsigned n)
{
    __shared__ v4f sc[VQ_K];  // 4 KB codebook in LDS

    // --- Stage codebook into LDS via CDNA5 async copy (one b128 per lane,
    // 256 lanes x 16 B = 4 KB; tracked by ASYNCcnt, not LOADcnt). ---
    {
        const unsigned tid = threadIdx.x;
        // Low 32 bits of the generic pointer == LDS byte offset (ISA 10.2:
        // the LDS address is the pointer's low 32 bits).
        unsigned lds_off = (unsigned)(unsigned long long)(&sc[tid]);
        unsigned long long gaddr =
            (unsigned long long)(const void*)(centroids + (size_t)tid * 4);
        asm volatile("global_load_async_to_lds_b128 %0, %1, off"
                     :
                     : "v"(lds_off), "v"(gaddr)
                     : "memory");
        asm volatile("s_wait_asynccnt 0" ::: "memory");
    }
    __syncthreads();

    const unsigned block_base = blockIdx.x * (VQ_BLOCK * VQ_UNROLL);
    const unsigned base       = block_base + threadIdx.x;

    if (block_base + VQ_BLOCK * VQ_UNROLL <= n) {
        // ---- Fast path: full block, no guards -> loads cluster into one
        // clause, gathers and stores overlap. Every instruction is fully
        // coalesced in wave32 (32 lanes x 16 B = 512 contiguous B / store). ----
        int idx[VQ_UNROLL];
#pragma unroll
        for (int k = 0; k < VQ_UNROLL; ++k) {
            // NT load: 64 MB index stream is touched exactly once.
            idx[k] = __builtin_nontemporal_load(&assignments[base + k * VQ_BLOCK]);
        }
#pragma unroll
        for (int k = 0; k < VQ_UNROLL; ++k) {
            // ds_load_b128 gather; mask keeps the LDS access in-range.
            v4f c = sc[idx[k] & (VQ_K - 1)];
            // NT store: 256 MB output stream exceeds the 192 MB L2.
            __builtin_nontemporal_store(c, &out[base + k * VQ_BLOCK]);
        }
    } else {
        // ---- Tail path: only the final block takes this. ----
#pragma unroll
        for (int k = 0; k < VQ_UNROLL; ++k) {
            const unsigned i = base + k * VQ_BLOCK;
            if (i < n) {
                int idx = __builtin_nontemporal_load(&assignments[i]);
                v4f c = sc[idx & (VQ_K - 1)];
                __builtin_nontemporal_store(c, &out[i]);
            }
        }
    }
}

extern "C" void kernel_launch(void* const* d_in, const int* in_sizes, int n_in,
                              void* d_out, int out_size, void* d_ws, size_t ws_size,
                              hipStream_t stream) {
    (void)n_in; (void)out_size; (void)d_ws; (void)ws_size;

    const float* centroids   = (const float*)d_in[0];   // 256*4 f32
    const int*   assignments = (const int*)d_in[1];     // 16,777,216 indices
    v4f*         out         = (v4f*)d_out;             // one 16B slot per index

    const unsigned n = (unsigned)in_sizes[1];
    const unsigned per_block = VQ_BLOCK * VQ_UNROLL;            // 2048
    const unsigned blocks = (n + per_block - 1) / per_block;    // 8192

    Quantize_53017076302344_kernel<<<blocks, VQ_BLOCK, 0, stream>>>(
        centroids, assignments, out, n);
}